// MPOGNN_56891136803554
// MI455X (gfx1250) — compile-verified
//
#include <hip/hip_runtime.h>
#include <math.h>

typedef __attribute__((ext_vector_type(2))) float v2f;
typedef __attribute__((ext_vector_type(8))) float v8f;

#define HDIM 128

// ---------------- degree / normalization ----------------
__global__ void deg_init_kernel(float* deg, int N) {
  int i = blockIdx.x * blockDim.x + threadIdx.x;
  if (i < N) deg[i] = 1.0f;  // self-loop contribution
}

__global__ void deg_edge_kernel(float* deg, const int* __restrict__ dst, int E) {
  int e = blockIdx.x * blockDim.x + threadIdx.x;
  if (e < E) atomicAdd(&deg[dst[e]], 1.0f);
}

__global__ void deg_fin_kernel(float* deg, int N) {
  int i = blockIdx.x * blockDim.x + threadIdx.x;
  if (i < N) {
    float d = deg[i];
    deg[i] = d > 0.0f ? rsqrtf(d) : 0.0f;
  }
}

// ---------------- fp32 WMMA GEMM: out = A[M,128] @ W[NCOLS,128]^T ----------------
// EPI==0: v *= dis[row]; store to out0 and out1 (acc init == self-loop term)
// EPI==1: v += bias[col]; relu; store to out0
template <int NCOLS, int EPI>
__global__ __launch_bounds__(NCOLS * 2)
void gemm_f32_wmma_kernel(const float* __restrict__ A,
                          const float* __restrict__ W,
                          const float* __restrict__ dis,
                          const float* __restrict__ bias,
                          float* __restrict__ out0,
                          float* __restrict__ out1,
                          int M) {
  constexpr int K = HDIM;
  constexpr int KSTEPS = K / 4;
  // A strip staged transposed: sAt[k*17 + m], stride 17 => bank-conflict-free
  __shared__ float sAt[K * 17 + 16];

  const int tid = threadIdx.x;
  const int wave = tid >> 5;          // column tile
  const int lane = tid & 31;
  const int idx15 = lane & 15;        // M index for A-frag, N index for B/C-frag
  const int kb = (lane >> 4) << 1;    // K sub-offset (0 or 2)
  const int col = wave * 16 + idx15;  // output column

  // Preload this wave's B fragments (W^T chunk) into registers: 64 VGPRs.
  v2f bfrag[KSTEPS];
#pragma unroll
  for (int s = 0; s < KSTEPS; ++s) {
    const float* wp = W + (size_t)col * K + s * 4 + kb;
    bfrag[s].x = wp[0];
    bfrag[s].y = wp[1];
  }
  float bias_v = 0.0f;
  if (EPI == 1) bias_v = bias[col];

  const int mtiles = M >> 4;  // M is a multiple of 16
  for (int mt = blockIdx.x; mt < mtiles; mt += gridDim.x) {
    __syncthreads();
    // Coalesced load of 16x128 A strip, stored transposed into LDS.
#pragma unroll
    for (int i = tid; i < 16 * K; i += NCOLS * 2) {
      int r = i >> 7;
      int c = i & (K - 1);
      sAt[c * 17 + r] = A[((size_t)mt * 16 + r) * K + c];
    }
    __syncthreads();

    v8f accv = {};
#pragma unroll
    for (int s = 0; s < KSTEPS; ++s) {
      const int k = s * 4 + kb;
      v2f a;
      a.x = sAt[k * 17 + idx15];
      a.y = sAt[(k + 1) * 17 + idx15];
      accv = __builtin_amdgcn_wmma_f32_16x16x4_f32(
          false, a, false, bfrag[s], (short)0, accv, false, false);
    }

    // C layout: lane holds rows mbase..mbase+7 of column `col`
    const int mbase = (lane >> 4) << 3;
#pragma unroll
    for (int i = 0; i < 8; ++i) {
      const size_t row = (size_t)mt * 16 + mbase + i;
      float v = accv[i];
      if (EPI == 0) {
        v *= dis[row];
        out0[row * NCOLS + col] = v;
        out1[row * NCOLS + col] = v;
      } else {
        v += bias_v;
        v = fmaxf(v, 0.0f);
        out0[row * NCOLS + col] = v;
      }
    }
  }
}

// ---------------- edge scatter-add (L2-resident atomics) ----------------
__global__ void scatter_kernel(const int* __restrict__ src,
                               const int* __restrict__ dst,
                               const float* __restrict__ hw,
                               float* __restrict__ acc, int E) {
  long long gid = (long long)blockIdx.x * blockDim.x + threadIdx.x;
  int e = (int)(gid >> 5);
  int lane = (int)(gid & 31);
  if (e >= E) return;
  int s = src[e];
  int d = dst[e];
  const float4* vp = (const float4*)(hw + (size_t)s * HDIM);
  float4 v = vp[lane];
  float* p = acc + (size_t)d * HDIM + lane * 4;
  atomicAdd(p + 0, v.x);
  atomicAdd(p + 1, v.y);
  atomicAdd(p + 2, v.z);
  atomicAdd(p + 3, v.w);
}

// ---------------- bias + residual dis scale + BN(eval) + ReLU ----------------
__global__ void bn_relu_kernel(const float* __restrict__ acc,
                               const float* __restrict__ dis,
                               const float* __restrict__ bias,
                               const float* __restrict__ gamma,
                               const float* __restrict__ beta,
                               const float* __restrict__ mean,
                               const float* __restrict__ var,
                               float* __restrict__ h, int N) {
  const int total = N * HDIM;
  const int stride = gridDim.x * blockDim.x;
  for (int i = blockIdx.x * blockDim.x + threadIdx.x; i < total; i += stride) {
    int c = i & (HDIM - 1);
    int n = i >> 7;
    float v = acc[i] * dis[n] + bias[c];
    v = (v - mean[c]) * rsqrtf(var[c] + 1e-5f) * gamma[c] + beta[c];
    h[i] = fmaxf(v, 0.0f);
  }
}

// ---------------- node head second layer: [N,64] -> [N,2] ----------------
__global__ void node_head2_kernel(const float* __restrict__ t,
                                  const float* __restrict__ w2,
                                  const float* __restrict__ b2,
                                  float* __restrict__ out, int N) {
  int n = blockIdx.x * blockDim.x + threadIdx.x;
  if (n >= N) return;
  const float* tp = t + (size_t)n * 64;
  float a0 = b2[0], a1 = b2[1];
#pragma unroll 16
  for (int k = 0; k < 64; ++k) {
    float tv = tp[k];
    a0 = fmaf(tv, w2[k], a0);
    a1 = fmaf(tv, w2[64 + k], a1);
  }
  out[(size_t)n * 2 + 0] = a0;
  out[(size_t)n * 2 + 1] = a1;
}

// ---------------- graph pooling (batch is sorted) ----------------
__global__ void pool_zero_kernel(float* gbuf, int total) {
  int i = blockIdx.x * blockDim.x + threadIdx.x;
  if (i < total) gbuf[i] = 0.0f;
}

__global__ void pool_block_kernel(const float* __restrict__ h,
                                  const int* __restrict__ batch,
                                  float* __restrict__ gsum,
                                  float* __restrict__ gcnt,
                                  int N) {
  __shared__ float lsum[64 * HDIM];  // 32KB
  __shared__ float lcnt[64];
  const int c = threadIdx.x;  // 0..127, owns column c (no LDS races)
  const int base = blockIdx.x * HDIM;
  for (int i = c; i < 64 * HDIM; i += HDIM) lsum[i] = 0.0f;
  if (c < 64) lcnt[c] = 0.0f;
  __syncthreads();
  int nmax = N - base;
  if (nmax > HDIM) nmax = HDIM;
  int curg = -1;
  float accv = 0.0f;
  for (int r = 0; r < nmax; ++r) {
    const int n = base + r;
    const int g = batch[n];
    if (g != curg) {
      if (curg >= 0) lsum[curg * HDIM + c] += accv;
      curg = g;
      accv = 0.0f;
    }
    accv += h[(size_t)n * HDIM + c];
    if (c == 0) lcnt[g] += 1.0f;
  }
  if (curg >= 0) lsum[curg * HDIM + c] += accv;
  __syncthreads();
  const int gmin = batch[base];
  const int gmax = batch[base + nmax - 1];
  for (int g = gmin; g <= gmax; ++g)
    atomicAdd(&gsum[g * HDIM + c], lsum[g * HDIM + c]);
  if (c == 0)
    for (int g = gmin; g <= gmax; ++g) atomicAdd(&gcnt[g], lcnt[g]);
}

// ---------------- graph head MLP (G=64 tiny) ----------------
__global__ void graph_head_kernel(const float* __restrict__ gsum,
                                  const float* __restrict__ gcnt,
                                  const float* __restrict__ w1,
                                  const float* __restrict__ b1,
                                  const float* __restrict__ w2,
                                  const float* __restrict__ b2,
                                  float* __restrict__ out) {
  __shared__ float emb[HDIM];
  __shared__ float t[64];
  const int g = blockIdx.x;
  const int j = threadIdx.x;  // 0..63
  const float inv = 1.0f / fmaxf(gcnt[g], 1.0f);
  for (int i = j; i < HDIM; i += 64) emb[i] = gsum[g * HDIM + i] * inv;
  __syncthreads();
  float a = b1[j];
#pragma unroll 16
  for (int k = 0; k < HDIM; ++k) a = fmaf(emb[k], w1[j * HDIM + k], a);
  t[j] = fmaxf(a, 0.0f);
  __syncthreads();
  if (j < 2) {
    float o = b2[j];
#pragma unroll 16
    for (int k = 0; k < 64; ++k) o = fmaf(t[k], w2[j * 64 + k], o);
    out[g * 2 + j] = o;
  }
}

extern "C" void kernel_launch(void* const* d_in, const int* in_sizes, int n_in,
                              void* d_out, int out_size, void* d_ws, size_t ws_size,
                              hipStream_t stream) {
  const float* x      = (const float*)d_in[0];
  const int*   eidx   = (const int*)d_in[1];
  const int*   batch  = (const int*)d_in[2];
  const float* conv_w = (const float*)d_in[3];
  const float* conv_b = (const float*)d_in[4];
  const float* bn_g   = (const float*)d_in[5];
  const float* bn_b   = (const float*)d_in[6];
  const float* bn_m   = (const float*)d_in[7];
  const float* bn_v   = (const float*)d_in[8];
  const float* nw1    = (const float*)d_in[9];
  const float* nb1    = (const float*)d_in[10];
  const float* nw2    = (const float*)d_in[11];
  const float* nb2    = (const float*)d_in[12];
  const float* gw1    = (const float*)d_in[13];
  const float* gb1    = (const float*)d_in[14];
  const float* gw2    = (const float*)d_in[15];
  const float* gb2    = (const float*)d_in[16];

  const int N = in_sizes[0] / HDIM;            // 100000
  const int E = in_sizes[1] / 2;               // 1600000
  const int L = in_sizes[3] / (HDIM * HDIM);   // 3
  const int G = (out_size - 2 * N) / 2;        // 64

  const int* src = eidx;
  const int* dst = eidx + E;

  // workspace layout (floats)
  float* dis  = (float*)d_ws;                  // N
  float* h    = dis + N;                       // N*128
  float* hw   = h + (size_t)N * HDIM;          // N*128
  float* acc  = hw + (size_t)N * HDIM;         // N*128
  float* tbuf = acc + (size_t)N * HDIM;        // N*64
  float* gsum = tbuf + (size_t)N * 64;         // G*128
  float* gcnt = gsum + (size_t)G * HDIM;       // G

  float* node_out  = (float*)d_out;
  float* graph_out = node_out + (size_t)N * 2;

  // symmetric normalization: dis[n] = 1/sqrt(deg[n]) (self-loops included)
  deg_init_kernel<<<(N + 255) / 256, 256, 0, stream>>>(dis, N);
  deg_edge_kernel<<<(E + 255) / 256, 256, 0, stream>>>(dis, dst, E);
  deg_fin_kernel<<<(N + 255) / 256, 256, 0, stream>>>(dis, N);

  const int mtiles = N / 16;
  const int gemm_grid = mtiles < 2048 ? mtiles : 2048;

  const float* hin = x;
  for (int l = 0; l < L; ++l) {
    // hw = (hin @ W^T) * dis[row]; acc initialized to hw (self-loop term)
    gemm_f32_wmma_kernel<HDIM, 0><<<gemm_grid, 256, 0, stream>>>(
        hin, conv_w + (size_t)l * HDIM * HDIM, dis, nullptr, hw, acc, N);
    // acc[dst] += hw[src] for all real edges (L2-resident atomics)
    long long sthreads = (long long)E * 32;
    scatter_kernel<<<(int)((sthreads + 255) / 256), 256, 0, stream>>>(
        src, dst, hw, acc, E);
    // h = relu(BN(acc*dis[dst] + b))
    bn_relu_kernel<<<4096, 256, 0, stream>>>(
        acc, dis, conv_b + l * HDIM, bn_g + l * HDIM, bn_b + l * HDIM,
        bn_m + l * HDIM, bn_v + l * HDIM, h, N);
    hin = h;
  }

  // node head: t = relu(h @ w1^T + b1); node_pred = t @ w2^T + b2
  gemm_f32_wmma_kernel<64, 1><<<gemm_grid, 128, 0, stream>>>(
      h, nw1, nullptr, nb1, tbuf, nullptr, N);
  node_head2_kernel<<<(N + 255) / 256, 256, 0, stream>>>(
      tbuf, nw2, nb2, node_out, N);

  // graph mean-pool + head
  pool_zero_kernel<<<(G * HDIM + G + 255) / 256, 256, 0, stream>>>(
      gsum, G * HDIM + G);
  pool_block_kernel<<<(N + HDIM - 1) / HDIM, HDIM, 0, stream>>>(
      h, batch, gsum, gcnt, N);
  graph_head_kernel<<<G, 64, 0, stream>>>(gsum, gcnt, gw1, gb1, gw2, gb2,
                                          graph_out);
}